// ProtoNet_53429393162420
// MI455X (gfx1250) — compile-verified
//
#include <hip/hip_runtime.h>

typedef float v2f __attribute__((ext_vector_type(2)));
typedef float v8f __attribute__((ext_vector_type(8)));

#define N_BL     8192
#define OFF_ZERO 16384
#define OFF_SP   16385
#define OFF_ST   81921
#define OFF_T    1523713
#define OFF_S    1769473
#define OFF_RES  8257537
#define OFF_LOSS 8258397

// ---------------------------------------------------------------------------
// Kernel A: merge (w*5x5 + (1-w)*3x3 or swapped) + directional pooling GEMM.
// One wave = one 16x16 WMMA tile: 16 (bl,ch) rows x 16 direction cols
// (11 valid), K = 49 padded to 52 -> 13 V_WMMA_F32_16X16X4_F32 steps.
// Border/zero-pad/branch-select all folded into per-k coefficient tables:
//   a[k] = v3[k]*c3[k] + v5[clamped_off[k]]*c5[k]
// so the inner loop is branch-free (no exec save/restore).
// ---------------------------------------------------------------------------
__global__ __launch_bounds__(256)
void pool_wmma_kernel(const float* __restrict__ src,
                      const float* __restrict__ masks,   // (11,9,9)
                      const float* __restrict__ locw,    // (49)
                      float* __restrict__ enc_out,       // (bl, 11, chs)
                      int chs, int w_on_5, int n_tiles)
{
    __shared__ float  msum[11];
    __shared__ float4 ck4[26];        // {c3(2i), c5(2i), c3(2i+1), c5(2i+1)}
    __shared__ int    offp[52];       // clamped v5 element offsets (0 if unused)
    __shared__ float2 bm2[26 * 16];   // {b(2i,n), b(2i+1,n)}

    const int tid = threadIdx.x;
    if (tid < 11) {
        float s = 0.f;
        for (int i = 0; i < 81; ++i) s += masks[tid * 81 + i];
        msum[tid] = s;
    }
    if (tid < 26) {
        float c[4];
        #pragma unroll
        for (int j = 0; j < 2; ++j) {
            int k = 2 * tid + j;
            float c3 = 0.f, c5 = 0.f;
            int off = 0;
            if (k < 49) {
                float w = 1.0f / (1.0f + __expf(-locw[k]));
                float cv5 = w_on_5 ? w : (1.0f - w);
                c3 = 1.0f - cv5;
                int x = k / 7, y = k - (k / 7) * 7;
                if (x >= 1 && x <= 5 && y >= 1 && y <= 5) {
                    c5 = cv5;
                    off = 49 + (x - 1) * 5 + (y - 1);
                }
            }
            c[2 * j] = c3;
            c[2 * j + 1] = c5;
            offp[k] = off;
        }
        ck4[tid] = make_float4(c[0], c[1], c[2], c[3]);
    }
    __syncthreads();
    for (int idx = tid; idx < 26 * 16; idx += 256) {
        int i = idx >> 4, n = idx & 15;
        float bv[2];
        #pragma unroll
        for (int j = 0; j < 2; ++j) {
            int k = 2 * i + j;
            float v = 0.f;
            if (k < 49 && n < 11) {
                int x = k / 7, y = k - (k / 7) * 7;
                v = masks[n * 81 + (x + 1) * 9 + (y + 1)] / msum[n];
            }
            bv[j] = v;
        }
        bm2[idx] = make_float2(bv[0], bv[1]);
    }
    __syncthreads();

    const int lane  = tid & 31;
    const int mloc  = lane & 15;
    const int hi    = lane >> 4;
    const int gwave = blockIdx.x * 8 + (tid >> 5);
    const int nwave = gridDim.x * 8;
    const int blstride = chs * 74;   // floats per bl in src

    for (int tile = gwave; tile < n_tiles; tile += nwave) {
        const int rowbase = tile << 4;
        const int q0 = rowbase / chs;
        const int r0 = rowbase - q0 * chs;
        int cm = r0 + mloc, bl = q0, ch = cm;
        if (cm >= chs) { ch = cm - chs; ++bl; }
        const float* p = src + (size_t)bl * blstride + (ch >> 2) * 296 + (ch & 3);

        v8f acc = {0.f, 0.f, 0.f, 0.f, 0.f, 0.f, 0.f, 0.f};
        #pragma unroll
        for (int s = 0; s < 13; ++s) {
            const int i  = (s << 1) + hi;   // k-pair index, k0 = 2*i
            const int k0 = i << 1;
            float4 c  = ck4[i];             // broadcast ds_load_b128
            float2 bb = bm2[i * 16 + mloc]; // ds_load_b64
            float v3a = p[k0 << 2];
            float v3b = p[(k0 + 1) << 2];
            float v5a = p[offp[k0] << 2];
            float v5b = p[offp[k0 + 1] << 2];
            float a0 = fmaf(v5a, c.y, v3a * c.x);
            float a1 = fmaf(v5b, c.w, v3b * c.z);
            v2f av = {a0, a1};
            v2f bv = {bb.x, bb.y};
            acc = __builtin_amdgcn_wmma_f32_16x16x4_f32(false, av, false, bv,
                                                        (short)0, acc, false, false);
        }

        if (mloc < 11) {
            #pragma unroll
            for (int v = 0; v < 8; ++v) {
                int d = v + (hi << 3);
                int cd = r0 + d, bd = q0;
                if (cd >= chs) { cd -= chs; ++bd; }
                enc_out[(size_t)bd * (11 * chs) + mloc * chs + cd] = acc[v];
            }
        }
    }
}

// ---------------------------------------------------------------------------
// Kernel B: sp_raw = 1 / cdist(feats, protos) via Gram GEMM.
// One wave = one 16-row tile; K = 998 padded to 1000 -> 250 WMMA steps.
// Protos staged in LDS (float2-packed). Feature gather uses branchless
// address selection (cndmask, no exec divergence).
// ---------------------------------------------------------------------------
__device__ __forceinline__ const float* feataddr(const float* __restrict__ T,
                                                 const float* __restrict__ S,
                                                 const float* __restrict__ ST,
                                                 int row, int f)
{
    const float* pS  = S  + (size_t)row * 792 + (f - 30);
    const float* pST = ST + (size_t)row * 176 + (f - 822);
    const float* pT  = T  + (row >> 8) * 30 + f;
    const float* r = (f < 822) ? pS : pST;   // f in [998,1000) stays inside d_out
    return (f < 30) ? pT : r;
}

__global__ __launch_bounds__(256)
void cdist_wmma_kernel(const float* __restrict__ temporal,
                       const float* __restrict__ protos,
                       const float* __restrict__ S_enc,
                       const float* __restrict__ ST_enc,
                       float* __restrict__ sp_out)
{
    __shared__ float2 pl2[8 * 500];   // protos padded to K=1000, float2-packed
    __shared__ float  pn[16];

    const int tid = threadIdx.x;
    for (int idx = tid; idx < 4000; idx += 256) {
        int p = idx / 500, i = idx - p * 500;
        int k = 2 * i;
        float a = (k     < 998) ? protos[p * 998 + k]     : 0.f;
        float b = (k + 1 < 998) ? protos[p * 998 + k + 1] : 0.f;
        pl2[idx] = make_float2(a, b);
    }
    __syncthreads();
    if (tid < 16) {
        float s = 0.f;
        if (tid < 8)
            for (int i = 0; i < 500; ++i) {
                float2 v = pl2[tid * 500 + i];
                s = fmaf(v.x, v.x, s);
                s = fmaf(v.y, v.y, s);
            }
        pn[tid] = s;
    }
    __syncthreads();

    const int lane  = tid & 31;
    const int mloc  = lane & 15;
    const int hi    = lane >> 4;
    const int khalf = hi << 1;
    const int tile  = blockIdx.x * 8 + (tid >> 5);   // exactly 512 tiles
    const int row   = (tile << 4) + mloc;
    const int pbase = (mloc < 8 ? mloc : 0) * 500;   // clamped, value-masked below
    const bool valcol = (mloc < 8);

    v8f acc = {0.f, 0.f, 0.f, 0.f, 0.f, 0.f, 0.f, 0.f};
    float sq = 0.f;
    for (int s = 0; s < 250; ++s) {
        const int k0 = (s << 2) + khalf;
        const int k1 = k0 + 1;
        float a0 = *feataddr(temporal, S_enc, ST_enc, row, k0);
        float a1 = *feataddr(temporal, S_enc, ST_enc, row, k1);
        a0 = (k0 < 998) ? a0 : 0.f;
        a1 = (k1 < 998) ? a1 : 0.f;
        float2 bb = pl2[pbase + (k0 >> 1)];
        float b0 = valcol ? bb.x : 0.f;
        float b1 = valcol ? bb.y : 0.f;
        sq = fmaf(a0, a0, sq);
        sq = fmaf(a1, a1, sq);
        v2f av = {a0, a1};
        v2f bv = {b0, b1};
        acc = __builtin_amdgcn_wmma_f32_16x16x4_f32(false, av, false, bv,
                                                    (short)0, acc, false, false);
    }

    float rn = sq + __shfl_xor(sq, 16, 32);
    float rnv[8];
    #pragma unroll
    for (int v = 0; v < 8; ++v) rnv[v] = __shfl(rn, (hi << 3) + v, 32);

    if (mloc < 8) {
        float pnn = pn[mloc];
        #pragma unroll
        for (int v = 0; v < 8; ++v) {
            int r = (tile << 4) + (hi << 3) + v;
            float d2 = rnv[v] + pnn - 2.0f * acc[v];
            float dist = sqrtf(fmaxf(d2, 0.f));
            sp_out[(size_t)r * 8 + mloc] = 1.0f / dist;
        }
    }
}

// ---------------------------------------------------------------------------
// Kernel C: per-row head. Read raw sp, min/max-normalize in place, fc+softmax.
// ---------------------------------------------------------------------------
__global__ __launch_bounds__(256)
void head_kernel(const float* __restrict__ fcw, float* __restrict__ out)
{
    int row = blockIdx.x * 256 + threadIdx.x;
    if (row >= N_BL) return;
    float* sp = out + OFF_SP + (size_t)row * 8;
    float v[8];
    float mn = 3.4e38f;
    #pragma unroll
    for (int i = 0; i < 8; ++i) { v[i] = sp[i]; mn = fminf(mn, v[i]); }
    float mx = -3.4e38f;
    #pragma unroll
    for (int i = 0; i < 8; ++i) { v[i] -= mn; mx = fmaxf(mx, v[i]); }
    float l0 = 0.f, l1 = 0.f;
    #pragma unroll
    for (int i = 0; i < 8; ++i) {
        v[i] /= mx;
        sp[i] = v[i];
        l0 = fmaf(v[i], fcw[i], l0);
        l1 = fmaf(v[i], fcw[8 + i], l1);
    }
    float m  = fmaxf(l0, l1);
    float e0 = __expf(l0 - m), e1 = __expf(l1 - m);
    float inv = 1.0f / (e0 + e1);
    out[(size_t)row * 2 + 0] = e0 * inv;
    out[(size_t)row * 2 + 1] = e1 * inv;
}

// ---------------------------------------------------------------------------
// Kernel D: T_d broadcast + scalar zero + res ones.
// ---------------------------------------------------------------------------
__global__ __launch_bounds__(256)
void misc_kernel(const float* __restrict__ temporal, float* __restrict__ out)
{
    int i = blockIdx.x * 256 + threadIdx.x;
    if (i < N_BL * 30) {
        int row = i / 30;
        int t = i - row * 30;
        out[OFF_T + i] = temporal[(row >> 8) * 30 + t];
    }
    if (i < 860) out[OFF_RES + i] = 1.0f;
    if (i == 0)  out[OFF_ZERO] = 0.0f;
}

// ---------------------------------------------------------------------------
// Kernel E: prototype diversity loss: -sum(pairwise dist)/(64-8).
// ---------------------------------------------------------------------------
__global__ void loss_kernel(const float* __restrict__ protos, float* __restrict__ out)
{
    int tid = threadIdx.x;   // 32 threads, one wave
    float d = 0.f;
    if (tid < 28) {
        int i = 0, t = tid;
        while (t >= 7 - i) { t -= 7 - i; ++i; }
        int j = i + 1 + t;
        float s = 0.f;
        for (int k = 0; k < 998; ++k) {
            float dd = protos[i * 998 + k] - protos[j * 998 + k];
            s = fmaf(dd, dd, s);
        }
        d = sqrtf(fmaxf(s, 0.f));
    }
    #pragma unroll
    for (int o = 16; o > 0; o >>= 1) d += __shfl_xor(d, o, 32);
    if (tid == 0) out[OFF_LOSS] = -d / 28.0f;   // -2*sum_{i<j}/56
}

extern "C" void kernel_launch(void* const* d_in, const int* in_sizes, int n_in,
                              void* d_out, int out_size, void* d_ws, size_t ws_size,
                              hipStream_t stream) {
    const float* stv      = (const float*)d_in[0];  // (32,256,1184)
    const float* sv       = (const float*)d_in[1];  // (32,256,5328)
    const float* temporal = (const float*)d_in[2];  // (32,30)
    const float* masks    = (const float*)d_in[3];  // (11,9,9)
    const float* locw     = (const float*)d_in[4];  // (49)
    const float* protos   = (const float*)d_in[5];  // (8,998)
    const float* fcw      = (const float*)d_in[6];  // (2,8)
    float* out = (float*)d_out;

    // Spatial branch: chs=72, w on 5x5 part; tiles = 8192*72/16 = 36864
    pool_wmma_kernel<<<1024, 256, 0, stream>>>(sv, masks, locw, out + OFF_S, 72, 1, 36864);
    // Spatio-temporal branch: chs=16, w on 3x3 part; tiles = 8192*16/16 = 8192
    pool_wmma_kernel<<<256, 256, 0, stream>>>(stv, masks, locw, out + OFF_ST, 16, 0, 8192);
    // T_d / zero / res
    misc_kernel<<<(N_BL * 30 + 255) / 256, 256, 0, stream>>>(temporal, out);
    // sp_raw (reads S_enc/ST_enc written above, stream-ordered)
    cdist_wmma_kernel<<<64, 256, 0, stream>>>(temporal, protos, out + OFF_S, out + OFF_ST,
                                              out + OFF_SP);
    // normalize sp + fc + softmax
    head_kernel<<<N_BL / 256, 256, 0, stream>>>(fcw, out);
    // diversity loss
    loss_kernel<<<1, 32, 0, stream>>>(protos, out);
}